// SourcePE_40759239639759
// MI455X (gfx1250) — compile-verified
//
#include <hip/hip_runtime.h>
#include <cstdint>

// Problem geometry (fixed by the reference).
#define EMB    512
#define NBOX   1024
#define BATCH  64
#define D4     128                       // EMB/4
#define TOTAL  33554432u                 // NBOX*BATCH*EMB
#define HALF   16777216u                 // TOTAL/2  (threefry counter split point)
#define TBL_COORDS 1025                  // coords are in [0, 1024]
#define TBL_ELEMS  (TBL_COORDS * D4)     // 131200 floats
#define TBL_BYTES  ((size_t)TBL_ELEMS * 4)

// keep iff uniform < 0.9f.  uniform = m*2^-23 exactly (m = bits>>9) and
// 0.9f == 7549747*2^-23 exactly, so keep <=> bits < (7549747 << 9).
#define KEEP_THRESH 0xE6666600u

// ---------------------------------------------------------------------------
// den[j] = exp(-j * ln(10000) / 512 / 4)   (torch/JAX precedence, all f32)
// ---------------------------------------------------------------------------
__device__ __forceinline__ float pe_den(unsigned j) {
    float t = -(float)j * 9.210340371976184f;  // ln(10000) rounded to f32
    t = t / 512.0f;
    t = t / 4.0f;
    return expf(t);
}

// ---------------------------------------------------------------------------
// Bit-exact JAX threefry2x32 for key = jax.random.key(42) -> (k0=0, k1=42).
// ---------------------------------------------------------------------------
__device__ __forceinline__ uint32_t rotl32(uint32_t x, uint32_t r) {
    return (x << r) | (x >> (32u - r));
}

__device__ __forceinline__ void threefry2x32_42(uint32_t& x0, uint32_t& x1) {
    const uint32_t k0 = 0u, k1 = 42u;
    const uint32_t k2 = k0 ^ k1 ^ 0x1BD11BDAu;
    x0 += k0; x1 += k1;
#define TF_R(r) { x0 += x1; x1 = rotl32(x1, r); x1 ^= x0; }
    TF_R(13u) TF_R(15u) TF_R(26u) TF_R(6u)   x0 += k1; x1 += k2 + 1u;
    TF_R(17u) TF_R(29u) TF_R(16u) TF_R(24u)  x0 += k2; x1 += k0 + 2u;
    TF_R(13u) TF_R(15u) TF_R(26u) TF_R(6u)   x0 += k0; x1 += k1 + 3u;
    TF_R(17u) TF_R(29u) TF_R(16u) TF_R(24u)  x0 += k1; x1 += k2 + 4u;
    TF_R(13u) TF_R(15u) TF_R(26u) TF_R(6u)   x0 += k2; x1 += k0 + 5u;
#undef TF_R
}

// Integer-domain dropout: 1 compare + 1 cndmask per element, bit-exact vs JAX.
__device__ __forceinline__ float apply_drop(uint32_t bits, float x) {
    return (bits < KEEP_THRESH) ? x * (1.0f / 0.9f) : 0.0f;
}

// ---------------------------------------------------------------------------
// Kernel 1: tbl[v*128 + j] = sin(v * den[j]).  (x_pe == y_pe: MAXW == MAXH.)
// 525 KB -> L2 resident; removes 33.5M large-argument sinf from the hot loop.
// ---------------------------------------------------------------------------
__global__ void build_pe_table(float* __restrict__ tbl) {
    unsigned i = blockIdx.x * blockDim.x + threadIdx.x;
    if (i >= TBL_ELEMS) return;
    unsigned v = i >> 7;       // coordinate value 0..1024
    unsigned j = i & 127u;     // frequency index
    tbl[i] = sinf((float)v * pe_den(j));
}

// ---------------------------------------------------------------------------
// Kernel 2: fused PE-add + threefry dropout.
// Thread t owns float4 at flat base = 4t (lower half) and the mirrored float4
// at base+HALF (same b,e; n -> n+512): one threefry hash per element pair.
// Embedding chunks are staged via gfx1250 async-to-LDS so the global stream
// overlaps the ~300-op ARX chain; box int4s go through SMEM (wave-uniform).
// ---------------------------------------------------------------------------
template <bool USE_TABLE>
__global__ __launch_bounds__(256)
void source_pe_kernel(const float* __restrict__ emb,
                      const int*   __restrict__ boxes,
                      const float* __restrict__ tbl,
                      float*       __restrict__ out) {
    __shared__ __align__(16) float tile[2048];   // 4 KB lower + 4 KB upper
    const unsigned tid  = threadIdx.x;
    const unsigned base = (blockIdx.x * 256u + tid) * 4u;   // < HALF, grid-exact

    // ---- issue async embedding fetches (ASYNCcnt path) --------------------
    const uint32_t lds_lo = (uint32_t)(uintptr_t)&tile[tid * 4u];
    const uint32_t lds_hi = lds_lo + 4096u;
    const uint32_t off_lo = base * 4u;            // byte offset, < 2^27
    const uint32_t off_hi = off_lo + HALF * 4u;   // < 2^28
    asm volatile("global_load_async_to_lds_b128 %0, %1, %2"
                 :: "v"(lds_lo), "v"(off_lo), "s"(emb) : "memory");
    asm volatile("global_load_async_to_lds_b128 %0, %1, %2"
                 :: "v"(lds_hi), "v"(off_hi), "s"(emb) : "memory");

    // ---- wave-uniform row -> scalar (SMEM) box loads ----------------------
    // flat idx = (n*64 + b)*512 + e, so row = base>>9 = n*64 + b, uniform
    // across each wave (a wave spans 128 consecutive e within one row).
    const unsigned row = __builtin_amdgcn_readfirstlane(base >> 9);
    const unsigned n   = row >> 6;             // 0..511 (lower half)
    const unsigned b   = row & 63u;
    const unsigned j   = (base >> 2) & 127u;   // per-lane frequency index
    const int4 c_lo = *(const int4*)(boxes + (((b << 10) + n)        << 2));
    const int4 c_hi = *(const int4*)(boxes + (((b << 10) + n + 512u) << 2));

    // ---- positional encoding ---------------------------------------------
    float pe_lo[4], pe_hi[4];
    if (USE_TABLE) {
        pe_lo[0] = tbl[(unsigned)c_lo.x * 128u + j];
        pe_lo[1] = tbl[(unsigned)c_lo.y * 128u + j];
        pe_lo[2] = tbl[(unsigned)c_lo.z * 128u + j];
        pe_lo[3] = tbl[(unsigned)c_lo.w * 128u + j];
        pe_hi[0] = tbl[(unsigned)c_hi.x * 128u + j];
        pe_hi[1] = tbl[(unsigned)c_hi.y * 128u + j];
        pe_hi[2] = tbl[(unsigned)c_hi.z * 128u + j];
        pe_hi[3] = tbl[(unsigned)c_hi.w * 128u + j];
    } else {
        const float den = pe_den(j);
        pe_lo[0] = sinf((float)c_lo.x * den);
        pe_lo[1] = sinf((float)c_lo.y * den);
        pe_lo[2] = sinf((float)c_lo.z * den);
        pe_lo[3] = sinf((float)c_lo.w * den);
        pe_hi[0] = sinf((float)c_hi.x * den);
        pe_hi[1] = sinf((float)c_hi.y * den);
        pe_hi[2] = sinf((float)c_hi.z * den);
        pe_hi[3] = sinf((float)c_hi.w * den);
    }

    // ---- 4 independent threefry chains (VALU ILP) -------------------------
    uint32_t rl[4], rh[4];
#pragma unroll
    for (int k = 0; k < 4; ++k) {
        uint32_t x0 = base + (unsigned)k;     // counter i       (lower elem)
        uint32_t x1 = x0 + HALF;              // counter i+half  (upper elem)
        threefry2x32_42(x0, x1);
        rl[k] = x0; rh[k] = x1;
    }

    // ---- drain async copies, combine, store -------------------------------
    asm volatile("s_wait_asynccnt 0" ::: "memory");
    const float4 vlo = *(const float4*)&tile[tid * 4u];
    const float4 vhi = *(const float4*)&tile[1024u + tid * 4u];

    float4 olo, ohi;
    olo.x = apply_drop(rl[0], vlo.x + pe_lo[0]);
    olo.y = apply_drop(rl[1], vlo.y + pe_lo[1]);
    olo.z = apply_drop(rl[2], vlo.z + pe_lo[2]);
    olo.w = apply_drop(rl[3], vlo.w + pe_lo[3]);
    ohi.x = apply_drop(rh[0], vhi.x + pe_hi[0]);
    ohi.y = apply_drop(rh[1], vhi.y + pe_hi[1]);
    ohi.z = apply_drop(rh[2], vhi.z + pe_hi[2]);
    ohi.w = apply_drop(rh[3], vhi.w + pe_hi[3]);

    *(float4*)(out + base)        = olo;
    *(float4*)(out + base + HALF) = ohi;
}

extern "C" void kernel_launch(void* const* d_in, const int* in_sizes, int n_in,
                              void* d_out, int out_size, void* d_ws, size_t ws_size,
                              hipStream_t stream) {
    const float* emb   = (const float*)d_in[0];   // (1024, 64, 512) f32
    const int*   boxes = (const int*)  d_in[1];   // (64, 1024, 4)   i32
    float*       out   = (float*)d_out;           // (1024, 64, 512) f32

    const unsigned main_blocks = HALF / (256u * 4u);   // 16384, grid-exact

    if (ws_size >= TBL_BYTES) {
        float* tbl = (float*)d_ws;
        build_pe_table<<<(TBL_ELEMS + 255) / 256, 256, 0, stream>>>(tbl);
        source_pe_kernel<true><<<main_blocks, 256, 0, stream>>>(emb, boxes, tbl, out);
    } else {
        source_pe_kernel<false><<<main_blocks, 256, 0, stream>>>(emb, boxes, nullptr, out);
    }
}